// FilterBank_20624432955781
// MI455X (gfx1250) — compile-verified
//
#include <hip/hip_runtime.h>

// FIR band-pass (413 taps, zero-phase, reflect-limited padding) as im2col GEMM
// on the CDNA5 matrix pipe using V_WMMA_F32_16X16X4_F32 (exact fp32 math).

typedef float v2f __attribute__((ext_vector_type(2)));
typedef float v8f __attribute__((ext_vector_type(8)));

#define T_LEN   10000
#define N_H     413
#define N_EDGE  412                         // N_H - 1 (== min(N_H, T) - 1)
#define EXT_LEN (T_LEN + 2 * N_EDGE)        // 10824
#define SHIFT   ((N_H - 1) / 2)             // 206 (zero-phase delay)
#define K_TOT   (N_H + 15)                  // 428, multiple of 4
#define K_STEPS (K_TOT / 4)                 // 107 WMMA steps
#define WIN     (16 * 15 + K_TOT)           // 668 floats of x_ext per tile
#define TILES   40                          // ceil(10000 / 256)
#define WAVES   8

// x_ext (reflect-limited padded signal) element i in [0, EXT_LEN)
__device__ __forceinline__ float xext_load(const float* __restrict__ xr, int i) {
    if (i > EXT_LEN - 1) i = EXT_LEN - 1;   // clamp (only hits padding of invalid tail outputs)
    if (i < N_EDGE) {
        return 2.0f * xr[0] - xr[N_EDGE - i];
    } else if (i < N_EDGE + T_LEN) {
        return xr[i - N_EDGE];
    } else {
        return 2.0f * xr[T_LEN - 1] - xr[2 * T_LEN - 2 - (i - N_EDGE)];
    }
}

__global__ __launch_bounds__(256) void fir_wmma_f32(const float* __restrict__ x,
                                                    const float* __restrict__ h,
                                                    float* __restrict__ out) {
    // Banded filter matrix B[k][j] = g[k-j] = h[412-(k-j)], stored pair-interleaved:
    // ldsB[(k>>1)*32 + j*2 + (k&1)]  so each lane's (k, k+1) pair is one b64 load.
    __shared__ __align__(16) float ldsB[K_TOT * 16];
    // Per-wave private sliding-window slab of x_ext.
    __shared__ __align__(16) float ldsW[WAVES][WIN];

    const int tid  = threadIdx.x;
    const int wave = tid >> 5;
    const int lane = tid & 31;
    const int hi   = lane >> 4;     // half-wave select (K split / M split)
    const int ln   = lane & 15;     // M for A-frag, N for B-frag and C/D

    const int bc = blockIdx.x;      // 64*64 rows
    const float* __restrict__ xr = x + (size_t)bc * T_LEN;
    float* __restrict__ orow     = out + (size_t)bc * T_LEN;

    // ---- build B once per block ----
    for (int e = tid; e < K_TOT * 16; e += 256) {
        const int k = e >> 4;
        const int j = e & 15;
        const int d = k - j;
        const float v = (d >= 0 && d < N_H) ? h[(N_H - 1) - d] : 0.0f;
        ldsB[(k >> 1) * 32 + j * 2 + (k & 1)] = v;
    }
    __syncthreads();

    const v2f* __restrict__ bp = (const v2f*)ldsB;
    const v2f* __restrict__ wp = (const v2f*)&ldsW[wave][0];
    const int aoff = 8 * ln + hi;   // float2 index: (16*M + 2*hi)/2
    const int boff = 16 * hi + ln;  // float2 index: ((2*hi)*16 + N)/... pair q*16+N

    // wave w handles tiles w, w+8, ... : exactly 5 iterations for every wave
    for (int tile = wave; tile < TILES; tile += WAVES) {
        // ---- fill per-wave window: win[p] = x_ext[tile*256 + 206 + p] ----
        const int base = tile * 256 + SHIFT;
        #pragma unroll
        for (int it = 0; it < (WIN + 31) / 32; ++it) {
            const int p = lane + 32 * it;
            if (p < WIN) ldsW[wave][p] = xext_load(xr, base + p);
        }
        __syncthreads();   // equal trip counts per wave -> safe; orders fill vs. reads

        // ---- 16x16 output tile: acc = A(16x428) * B(428x16) ----
        v8f acc = {};
        #pragma unroll 4
        for (int s = 0; s < K_STEPS; ++s) {
            const v2f a = wp[aoff + 2 * s];       // A[M][4s+2hi .. +1]
            const v2f b = bp[boff + 32 * s];      // B[4s+2hi .. +1][N]
            acc = __builtin_amdgcn_wmma_f32_16x16x4_f32(
                false, a, false, b, (short)0, acc, false, false);
        }

        // ---- store: VGPR r, lane -> t = tile*256 + 16*(r + 8*hi) + N ----
        const int tb = tile * 256 + 128 * hi + ln;
        #pragma unroll
        for (int r = 0; r < 8; ++r) {
            const int t = tb + 16 * r;
            if (t < T_LEN) orow[t] = acc[r];
        }
        __syncthreads();   // protect window slab before next iteration's refill
    }
}

extern "C" void kernel_launch(void* const* d_in, const int* in_sizes, int n_in,
                              void* d_out, int out_size, void* d_ws, size_t ws_size,
                              hipStream_t stream) {
    const float* x = (const float*)d_in[0];   // [64, 64, 10000] fp32
    const float* h = (const float*)d_in[1];   // [413] fp32
    float* out = (float*)d_out;               // [64, 64, 10000] fp32
    (void)in_sizes; (void)n_in; (void)out_size; (void)d_ws; (void)ws_size;

    fir_wmma_f32<<<dim3(64 * 64), dim3(256), 0, stream>>>(x, h, out);
}